// WildCatPoolDecision_39410619908430
// MI455X (gfx1250) — compile-verified
//
#include <hip/hip_runtime.h>
#include <stdint.h>

// Problem constants from the reference: x is (32, 200, 112, 112) fp32.
#define N_PER_ROW 12544      // 112*112
#define KMAX      6272       // round(0.5 * 12544)
#define THREADS   256        // 8 wave32s
#define NWAVES    8
#define EPT       49         // elements per thread: 12544 / 256

// ---- CDNA5 async global->LDS helpers (ASYNCcnt path, ISA 15.18 op 96/98) ----
// Read-once stream -> non-temporal hint so 321 MB doesn't churn the caches.
__device__ __forceinline__ void async_copy_b128(uint32_t lds_byte_addr, uint64_t gaddr) {
  asm volatile("global_load_async_to_lds_b128 %0, %1, off th:TH_LOAD_NT"
               :: "v"(lds_byte_addr), "v"(gaddr) : "memory");
}
__device__ __forceinline__ void async_copy_b32(uint32_t lds_byte_addr, uint64_t gaddr) {
  asm volatile("global_load_async_to_lds_b32 %0, %1, off th:TH_LOAD_NT"
               :: "v"(lds_byte_addr), "v"(gaddr) : "memory");
}
__device__ __forceinline__ void wait_asynccnt0() {
  asm volatile("s_wait_asynccnt 0" ::: "memory");
}

__global__ __launch_bounds__(THREADS) void
wildcat_topk_mean_kernel(const float* __restrict__ x, float* __restrict__ out) {
  __shared__ float    smem[N_PER_ROW];   // 50176 B: one full row staged in LDS
  __shared__ uint32_t hist[256];
  __shared__ uint32_t waveTot[NWAVES];
  __shared__ uint32_t s_prefix;
  __shared__ uint32_t s_want;
  __shared__ float    s_partial[NWAVES];

  const uint32_t tid  = threadIdx.x;
  const uint32_t lane = tid & 31u;       // wave32
  const uint32_t wid  = tid >> 5;
  const uint64_t row  = blockIdx.x;
  const uint8_t* src  = (const uint8_t*)(x + row * (uint64_t)N_PER_ROW);

  // ---------------- Phase A: async DMA global -> LDS (single HBM pass) ------
  {
    // Generic pointer to LDS: low 32 bits are the LDS byte offset (ISA 10.2
    // aperture rule); same base the async op adds.
    const uint32_t lds_base = (uint32_t)(uintptr_t)(&smem[0]);
#pragma unroll
    for (int i = 0; i < 12; ++i) {                 // 12 x b128 = 12288 floats
      const uint32_t off = tid * 16u + (uint32_t)i * 4096u;
      async_copy_b128(lds_base + off, (uint64_t)(src + off));
    }
    {                                              // tail: 256 floats via b32
      const uint32_t off = 49152u + tid * 4u;
      async_copy_b32(lds_base + off, (uint64_t)(src + off));
    }
    wait_asynccnt0();     // this wave's async loads landed in LDS
  }
  __syncthreads();        // make all waves' DMA visible workgroup-wide

  // ---------------- Load my 49 elements once into registers as sort keys ----
  // Order-preserving map: f>=0 -> u|0x80000000 ; f<0 -> ~u
  uint32_t keys[EPT];
#pragma unroll
  for (int i = 0; i < EPT; ++i) {
    const uint32_t u = __float_as_uint(smem[tid + i * THREADS]); // conflict-free stride
    keys[i] = (u & 0x80000000u) ? ~u : (u | 0x80000000u);
  }

  // ---------------- Phase B: exact MSB-first radix select of k-th largest ---
  uint32_t prefix = 0;      // known high bits of the k-th largest key
  uint32_t want   = KMAX;   // how many top elements still to account for
#pragma unroll
  for (int pass = 0; pass < 4; ++pass) {
    const int      shift = 24 - pass * 8;
    const uint32_t mask  = (pass == 0) ? 0u : (0xFFFFFFFFu << (shift + 8));

    hist[tid] = 0u;
    __syncthreads();

#pragma unroll
    for (int i = 0; i < EPT; ++i) {
      const uint32_t k = keys[i];
      if ((k & mask) == prefix) atomicAdd(&hist[(k >> shift) & 255u], 1u);
    }
    __syncthreads();

    // Parallel suffix scan: S[t] = sum_{b >= t} hist[b].
    const uint32_t h = hist[tid];
    uint32_t s = h;                       // intra-wave suffix (Hillis-Steele)
#pragma unroll
    for (int d = 1; d < 32; d <<= 1) {
      const uint32_t o = __shfl_down(s, (unsigned)d, 32);
      if (lane + (uint32_t)d < 32u) s += o;
    }
    if (lane == 0) waveTot[wid] = s;      // wave's total (suffix at its lowest bin)
    __syncthreads();

    uint32_t off = 0;                     // totals of all higher-bin waves
#pragma unroll
    for (int w2 = 0; w2 < NWAVES; ++w2)
      if ((uint32_t)w2 > wid) off += waveTot[w2];

    const uint32_t S     = s + off;       // inclusive suffix sum at bin `tid`
    const uint32_t Snext = S - h;         // suffix sum at bin tid+1
    if (S >= want && Snext < want) {      // unique crossing bin (h > 0 here)
      s_prefix = prefix | (tid << shift);
      s_want   = want - Snext;            // ties at this digit still needed
    }
    __syncthreads();
    prefix = s_prefix;
    want   = s_want;
    // (next pass's two barriers order these reads before any re-write)
  }
  // Now: prefix == exact key T of the k-th largest value,
  //      want   == number of copies of T to include (>= 1).

  // ---------------- Phase C: sum elements strictly above T + ties -----------
  float psum = 0.0f;
#pragma unroll
  for (int i = 0; i < EPT; ++i) {
    const uint32_t k = keys[i];
    if (k > prefix) {
      const uint32_t u = (k & 0x80000000u) ? (k & 0x7FFFFFFFu) : ~k; // invert key map
      psum += __uint_as_float(u);
    }
  }

  // Deterministic reduction: wave32 shfl tree, then 8 wave partials.
#pragma unroll
  for (int d = 16; d > 0; d >>= 1) psum += __shfl_down(psum, (unsigned)d, 32);
  if (lane == 0) s_partial[wid] = psum;
  __syncthreads();

  if (tid == 0) {
    float total = 0.0f;
#pragma unroll
    for (int w = 0; w < NWAVES; ++w) total += s_partial[w];
    const uint32_t ut = (prefix & 0x80000000u) ? (prefix & 0x7FFFFFFFu) : ~prefix;
    total += __uint_as_float(ut) * (float)want;   // include ties at threshold
    out[row] = total / (float)KMAX;
  }
}

extern "C" void kernel_launch(void* const* d_in, const int* in_sizes, int n_in,
                              void* d_out, int out_size, void* d_ws, size_t ws_size,
                              hipStream_t stream) {
  (void)n_in; (void)d_ws; (void)ws_size;
  const float* x  = (const float*)d_in[0];
  float*      out = (float*)d_out;
  int rows = out_size > 0 ? out_size : (in_sizes[0] / N_PER_ROW);
  wildcat_topk_mean_kernel<<<rows, THREADS, 0, stream>>>(x, out);
}